// GCNConv_15006615733818
// MI455X (gfx1250) — compile-verified
//
#include <hip/hip_runtime.h>
#include <hip/hip_bf16.h>

#define N_NODES 100000
#define N_EDGES 1600000
#define DIM     128

typedef __attribute__((ext_vector_type(2))) float v2f;
typedef __attribute__((ext_vector_type(8))) float v8f;

// ---------------------------------------------------------------------------
// 1) out[n, d] = bias[d];  deg[n] = 0
// ---------------------------------------------------------------------------
__global__ void gcn_init_kernel(float* __restrict__ out,
                                float* __restrict__ deg,
                                const float* __restrict__ bias,
                                int n_out, int n_nodes) {
    int i = blockIdx.x * blockDim.x + threadIdx.x;
    if (i < n_out)   out[i] = bias[i & (DIM - 1)];
    if (i < n_nodes) deg[i] = 0.0f;
}

// ---------------------------------------------------------------------------
// 2) deg[src[e]] += 1  (hardware f32 atomic, L2-resident 400KB array)
// ---------------------------------------------------------------------------
__global__ void gcn_degree_kernel(const long long* __restrict__ ei,
                                  float* __restrict__ deg, int n_edges) {
    int e = blockIdx.x * blockDim.x + threadIdx.x;
    if (e < n_edges) {
        int s = (int)ei[e];  // row 0 of edge_index = src
        unsafeAtomicAdd(&deg[s], 1.0f);
    }
}

// ---------------------------------------------------------------------------
// 3) dinv[n] = deg>0 ? rsqrt(deg) : 0   (in place)
// ---------------------------------------------------------------------------
__global__ void gcn_rsqrt_kernel(float* __restrict__ deg, int n) {
    int i = blockIdx.x * blockDim.x + threadIdx.x;
    if (i < n) {
        float d = deg[i];
        deg[i] = (d > 0.0f) ? rsqrtf(d) : 0.0f;
    }
}

// ---------------------------------------------------------------------------
// 4) h = x @ W  via V_WMMA_F32_16X16X4_F32.
//    block = 256 threads (8 waves). Block b owns rows [16b, 16b+16).
//    Wave w owns output cols [16w, 16w+16). K-loop: 32 x wmma (K=4 each).
//    A frag (16x4 f32): lane<16 -> M=lane, {K=k0,k0+1}; lane>=16 -> {k0+2,k0+3}
//    B frag (4x16 f32): mirrored, lane index = N within tile.
//    C frag: VGPR r -> M = r + 8*(lane>=16), N = lane&15.
// ---------------------------------------------------------------------------
__global__ void gcn_gemm_wmma_kernel(const float* __restrict__ x,
                                     const float* __restrict__ w,
                                     float* __restrict__ h) {
    const int lane = threadIdx.x & 31;
    const int half = lane >> 4;      // 0 or 1
    const int lm   = lane & 15;
    const int nt   = threadIdx.x >> 5;  // wave id = col tile 0..7

    const int mrow = blockIdx.x * 16 + lm;   // A-row this lane reads
    const int ncol = nt * 16 + lm;           // B-col this lane reads

    const float* xrow = x + (size_t)mrow * DIM + 2 * half;          // K offset
    const float* wcol = w + (size_t)(2 * half) * DIM + ncol;

    v8f acc = {};
#pragma unroll 4
    for (int k0 = 0; k0 < DIM; k0 += 4) {
        v2f a, b;
        a.x = xrow[k0 + 0];                       // x[mrow][k0 + 2*half]
        a.y = xrow[k0 + 1];                       // x[mrow][k0 + 2*half + 1]
        b.x = wcol[(size_t)k0 * DIM];             // w[k0 + 2*half][ncol]
        b.y = wcol[(size_t)k0 * DIM + DIM];       // w[k0 + 2*half + 1][ncol]
        acc = __builtin_amdgcn_wmma_f32_16x16x4_f32(
            /*neg_a=*/false, a, /*neg_b=*/false, b,
            /*c_mod=*/(short)0, acc, /*reuse_a=*/false, /*reuse_b=*/false);
    }

    float* hp = h + (size_t)(blockIdx.x * 16 + half * 8) * DIM + nt * 16 + lm;
#pragma unroll
    for (int r = 0; r < 8; ++r)
        hp[(size_t)r * DIM] = acc[r];
}

// ---------------------------------------------------------------------------
// 5) out[dst] += dinv[src]*dinv[dst] * h[src]
//    One wave32 per edge; each lane owns a float4 (32*16B = 512B = one row).
//    h and out both fit in 192MB L2 -> atomics/gathers are L2-resident.
// ---------------------------------------------------------------------------
__global__ void gcn_scatter_kernel(const long long* __restrict__ ei,
                                   const float* __restrict__ h,
                                   const float* __restrict__ dinv,
                                   float* __restrict__ out, int n_edges) {
    const int lane = threadIdx.x & 31;
    const int e = blockIdx.x * (blockDim.x >> 5) + (threadIdx.x >> 5);
    if (e >= n_edges) return;   // wave-uniform

    const int s = (int)ei[e];
    const int d = (int)ei[n_edges + e];
    const float norm = dinv[s] * dinv[d];

    const float4 v =
        reinterpret_cast<const float4*>(h + (size_t)s * DIM)[lane];
    float* op = out + (size_t)d * DIM + lane * 4;
    unsafeAtomicAdd(op + 0, norm * v.x);
    unsafeAtomicAdd(op + 1, norm * v.y);
    unsafeAtomicAdd(op + 2, norm * v.z);
    unsafeAtomicAdd(op + 3, norm * v.w);
}

// ---------------------------------------------------------------------------
extern "C" void kernel_launch(void* const* d_in, const int* in_sizes, int n_in,
                              void* d_out, int out_size, void* d_ws, size_t ws_size,
                              hipStream_t stream) {
    const float*      x    = (const float*)d_in[0];
    const long long*  ei   = (const long long*)d_in[1];   // int64 [2, E]
    const float*      w    = (const float*)d_in[2];
    const float*      bias = (const float*)d_in[3];
    float*            out  = (float*)d_out;

    float* h   = (float*)d_ws;                     // N*128 floats (51.2 MB)
    float* deg = h + (size_t)N_NODES * DIM;        // N floats (doubles as dinv)

    const int n_out = N_NODES * DIM;

    // 1) init out with bias, zero deg
    gcn_init_kernel<<<(n_out + 255) / 256, 256, 0, stream>>>(
        out, deg, bias, n_out, N_NODES);

    // 2) degree over src endpoints
    gcn_degree_kernel<<<(N_EDGES + 255) / 256, 256, 0, stream>>>(
        ei, deg, N_EDGES);

    // 3) dinv = rsqrt(deg) with isolated-node guard (in place)
    gcn_rsqrt_kernel<<<(N_NODES + 255) / 256, 256, 0, stream>>>(
        deg, N_NODES);

    // 4) h = x @ W  (WMMA f32; 6250 row-tiles x 8 col-tiles per block)
    gcn_gemm_wmma_kernel<<<N_NODES / 16, 256, 0, stream>>>(x, w, h);

    // 5) edge-parallel gather/scale/scatter-add (one wave per edge)
    gcn_scatter_kernel<<<(N_EDGES + 7) / 8, 256, 0, stream>>>(
        ei, h, deg, out, N_EDGES);
}